// Quantizer_fp4_47665547051587
// MI455X (gfx1250) — compile-verified
//
#include <hip/hip_runtime.h>

typedef float v4f __attribute__((ext_vector_type(4)));

#define ROW_COLS 4096
#define BLOCK    256
#define CHUNKS   (ROW_COLS / (BLOCK * 4))   // 4 x b128 per thread per row

// Symmetric FP4 codebook: code[8..15] = {0, .5, 1, 1.5, 2, 3, 4, 6} (magnitudes),
// code[0..7] are the negatives (code[7] = -0.0). Nearest codeword =
// magnitude bucket k = #{t in {.25,.75,1.25,1.75,2.5,3.5,5} : |q| > t},
// sign from q. The reference's argmin always resolves the {-0.0,+0.0} tie to
// index 7 (-0.0), so the k==0 bucket gets a forced sign bit.
__device__ __forceinline__ float nearest_dequant(float q, const float* lmag, float s) {
    const float a = __builtin_fabsf(q);        // free |src| modifier on v_cmp
    int k = 0;
    k += (int)(a > 0.25f);
    k += (int)(a > 0.75f);
    k += (int)(a > 1.25f);
    k += (int)(a > 1.75f);
    k += (int)(a > 2.5f);
    k += (int)(a > 3.5f);
    k += (int)(a > 5.0f);
    const float r = lmag[k] * s;               // ds_load_b32 gather, 8-entry LUT, r >= 0
    unsigned sgn = __builtin_bit_cast(unsigned, q) & 0x80000000u;
    sgn |= (k == 0) ? 0x80000000u : 0u;        // zero bucket -> -0.0 like reference
    return __builtin_bit_cast(float, __builtin_bit_cast(unsigned, r) | sgn);
}

// One block (8 wave32s) per TWO 4096-float rows. Both rows are staged to LDS
// with the CDNA5 async global->LDS path; the in-order ASYNCcnt lets us compute
// row A while row B's DMA is still in flight (split s_wait_asynccnt).
__global__ void __launch_bounds__(BLOCK)
quant_fp4_row_kernel(const float* __restrict__ x,
                     const float* __restrict__ scale,
                     const float* __restrict__ code,
                     float* __restrict__ out,
                     int rows) {
    __shared__ __align__(16) float lbufA[ROW_COLS];
    __shared__ __align__(16) float lbufB[ROW_COLS];
    __shared__ float lmag[8];

    const int tid   = threadIdx.x;
    const int rowA  = (int)blockIdx.x * 2;
    const int rowB  = rowA + 1;
    const bool hasB = rowB < rows;

    if (tid < 8) lmag[tid] = code[8 + tid];    // positive half of codebook

    const float* xA = x + (unsigned long long)rowA * ROW_COLS;
    const float* xB = x + (unsigned long long)rowB * ROW_COLS;
    float* oA = out + (unsigned long long)rowA * ROW_COLS;
    float* oB = out + (unsigned long long)rowB * ROW_COLS;

    // Issue async 128-bit global->LDS copies: row A first, then row B.
    // Each thread writes (and later reads) exactly its own 16B x 4 slices.
#pragma unroll
    for (int c = 0; c < CHUNKS; ++c) {
        const int e = c * (BLOCK * 4) + tid * 4;
        unsigned laddr = (unsigned)(unsigned long long)(&lbufA[e]);
        unsigned long long gaddr = (unsigned long long)(const void*)(xA + e);
        asm volatile("global_load_async_to_lds_b128 %0, %1, off"
                     :: "v"(laddr), "v"(gaddr) : "memory");
    }
    if (hasB) {
#pragma unroll
        for (int c = 0; c < CHUNKS; ++c) {
            const int e = c * (BLOCK * 4) + tid * 4;
            unsigned laddr = (unsigned)(unsigned long long)(&lbufB[e]);
            unsigned long long gaddr = (unsigned long long)(const void*)(xB + e);
            asm volatile("global_load_async_to_lds_b128 %0, %1, off"
                         :: "v"(laddr), "v"(gaddr) : "memory");
        }
    }

    const float sA  = scale[rowA];
    const float rsA = 1.0f / sA;
    const float sB  = hasB ? scale[rowB] : 1.0f;
    const float rsB = 1.0f / sB;

    __syncthreads();                           // lmag visibility only

    // Row A resident once ASYNCcnt <= CHUNKS (async loads complete in order).
    asm volatile("s_wait_asynccnt 0x4" ::: "memory");
#pragma unroll
    for (int c = 0; c < CHUNKS; ++c) {
        const int e = c * (BLOCK * 4) + tid * 4;
        v4f v = *(const v4f*)&lbufA[e];        // ds_load_b128
        v4f r;
        r.x = nearest_dequant(v.x * rsA, lmag, sA);
        r.y = nearest_dequant(v.y * rsA, lmag, sA);
        r.z = nearest_dequant(v.z * rsA, lmag, sA);
        r.w = nearest_dequant(v.w * rsA, lmag, sA);
        __builtin_nontemporal_store(r, (v4f*)(oA + e));
    }

    if (hasB) {
        asm volatile("s_wait_asynccnt 0x0" ::: "memory");
#pragma unroll
        for (int c = 0; c < CHUNKS; ++c) {
            const int e = c * (BLOCK * 4) + tid * 4;
            v4f v = *(const v4f*)&lbufB[e];
            v4f r;
            r.x = nearest_dequant(v.x * rsB, lmag, sB);
            r.y = nearest_dequant(v.y * rsB, lmag, sB);
            r.z = nearest_dequant(v.z * rsB, lmag, sB);
            r.w = nearest_dequant(v.w * rsB, lmag, sB);
            __builtin_nontemporal_store(r, (v4f*)(oB + e));
        }
    } else {
        asm volatile("s_wait_asynccnt 0x0" ::: "memory");
    }
}

// Generic fallback for shapes other than cols==4096 (grid-stride, scalar).
__global__ void __launch_bounds__(BLOCK)
quant_fp4_generic_kernel(const float* __restrict__ x,
                         const float* __restrict__ scale,
                         const float* __restrict__ code,
                         float* __restrict__ out,
                         int cols, long long n) {
    __shared__ float lmag[8];
    if (threadIdx.x < 8) lmag[threadIdx.x] = code[8 + threadIdx.x];
    __syncthreads();

    long long i = (long long)blockIdx.x * blockDim.x + threadIdx.x;
    const long long stride = (long long)gridDim.x * blockDim.x;
    for (; i < n; i += stride) {
        const int row = (int)(i / cols);
        const float s = scale[row];
        out[i] = nearest_dequant(x[i] / s, lmag, s);
    }
}

extern "C" void kernel_launch(void* const* d_in, const int* in_sizes, int n_in,
                              void* d_out, int out_size, void* d_ws, size_t ws_size,
                              hipStream_t stream) {
    const float* x     = (const float*)d_in[0];
    const float* scale = (const float*)d_in[1];
    const float* code  = (const float*)d_in[2];
    float* out = (float*)d_out;

    const long long n = (long long)in_sizes[0];
    const int rows = in_sizes[1] > 0 ? in_sizes[1] : 1;
    const int cols = (int)(n / rows);

    if (cols == ROW_COLS && (long long)rows * ROW_COLS == n) {
        const int blocks = (rows + 1) / 2;     // two rows per block
        quant_fp4_row_kernel<<<blocks, BLOCK, 0, stream>>>(x, scale, code, out, rows);
    } else {
        long long want = (n + BLOCK - 1) / BLOCK;
        int blocks = (int)(want < 32768 ? want : 32768);
        if (blocks < 1) blocks = 1;
        quant_fp4_generic_kernel<<<blocks, BLOCK, 0, stream>>>(x, scale, code, out, cols, n);
    }
}